// ConstrainModule_21466246545874
// MI455X (gfx1250) — compile-verified
//
#include <hip/hip_runtime.h>
#include <hip/hip_bf16.h>
#include <math.h>

// Problem constants from the reference
#define N_OBJ   1024
#define C_DET   8
#define C_SEG   4
#define HW      784          // 28*28
#define HW4     196          // HW / 4 (float4 tiles)
#define NBLK    (N_OBJ * C_DET)   // 8192 (n,j) pairs

typedef float v2f __attribute__((ext_vector_type(2)));
typedef float v8f __attribute__((ext_vector_type(8)));

// ---------------------------------------------------------------------------
// Stage 1: per-(n,j) block computes 4 dot products seg[n,j,i,:]·det[n,j,:]
// and the det tile sum. Pure streaming: b128 coalesced loads, LDS tree
// reduction, deterministic partials to workspace (no float atomics).
// partials layout: partials[q * NBLK + blk], q in {0..3}=A_i, 4=denom.
// ---------------------------------------------------------------------------
__global__ __launch_bounds__(256)
void cm_stage1_reduce(const float* __restrict__ det,
                      const float* __restrict__ seg,
                      float* __restrict__ partials) {
    const int blk = blockIdx.x;                 // blk = n*C_DET + j
    const int tid = threadIdx.x;

    const float4* d4 = (const float4*)(det + (size_t)blk * HW);
    const float4* s4 = (const float4*)(seg + (size_t)blk * (size_t)(C_SEG * HW));

    float acc[5] = {0.f, 0.f, 0.f, 0.f, 0.f};
    if (tid < HW4) {
        float4 d = d4[tid];
#pragma unroll
        for (int i = 0; i < C_SEG; ++i) {
            float4 s = s4[i * HW4 + tid];
            acc[i] = fmaf(s.x, d.x, fmaf(s.y, d.y, fmaf(s.z, d.z, s.w * d.w)));
        }
        acc[4] = (d.x + d.y) + (d.z + d.w);
    }

    __shared__ float red[256];
#pragma unroll
    for (int q = 0; q < 5; ++q) {
        red[tid] = acc[q];
        __syncthreads();
        for (int off = 128; off > 0; off >>= 1) {
            if (tid < off) red[tid] += red[tid + off];
            __syncthreads();
        }
        if (tid == 0) partials[q * NBLK + blk] = red[0];
        __syncthreads();
    }
}

// ---------------------------------------------------------------------------
// Stage 2: fold 5*8192 partials -> A[j][i], denom[j]; apply edge list to
// produce w[j] = sum_{edges (i,j)} A[j][i]/denom[j]. Single block,
// deterministic tree reductions.
// ---------------------------------------------------------------------------
__global__ __launch_bounds__(256)
void cm_stage2_edges(const float* __restrict__ partials,
                     const int* __restrict__ edge_i,
                     const int* __restrict__ edge_j,
                     int n_edges,
                     float* __restrict__ w_out) {
    const int tid = threadIdx.x;
    __shared__ float red[256];
    __shared__ float vals[5 * C_DET];           // vals[q*C_DET + j]

    for (int q = 0; q < 5; ++q) {
        for (int j = 0; j < C_DET; ++j) {
            float s = 0.f;
            for (int n = tid; n < N_OBJ; n += 256)
                s += partials[q * NBLK + n * C_DET + j];
            red[tid] = s;
            __syncthreads();
            for (int off = 128; off > 0; off >>= 1) {
                if (tid < off) red[tid] += red[tid + off];
                __syncthreads();
            }
            if (tid == 0) vals[q * C_DET + j] = red[0];
            __syncthreads();
        }
    }

    if (tid == 0) {
        float wacc[C_DET];
#pragma unroll
        for (int j = 0; j < C_DET; ++j) wacc[j] = 0.f;
        for (int e = 0; e < n_edges; ++e) {
            int i = edge_i[e];
            int j = edge_j[e];
            wacc[j] += vals[i * C_DET + j] / vals[4 * C_DET + j];
        }
#pragma unroll
        for (int j = 0; j < C_DET; ++j) w_out[j] = wacc[j];
    }
}

// ---------------------------------------------------------------------------
// Stage 3: probs = det_class_probs (1024x8) @ w (8), then
// bce = mean(-clip(log(probs), -100)). The matvec is done with exact-f32
// V_WMMA_F32_16X16X4_F32: per 16-object tile, two chained K=4 WMMAs.
//   A (16x4 f32): lanes 0-15 hold {K0,K1} of row M=lane, lanes 16-31 hold
//                 {K2,K3} of row M=lane-16.
//   B (4x16 f32): lanes 0-15 hold rows {K0,K1}, lanes 16-31 rows {K2,K3};
//                 B[k][n] = w[k] is n-uniform -> simple lane-half select.
//   D (16x16 f32): column N=0 lives in lane 0 (M=vgpr) and lane 16 (M=8+vgpr).
// One block of 512 threads (16 waves), each wave does 4 tiles of 16 objects.
// ---------------------------------------------------------------------------
__global__ __launch_bounds__(512)
void cm_stage3_wmma_bce(const float* __restrict__ dcp,
                        const float* __restrict__ w,
                        float* __restrict__ out) {
    const int tid  = threadIdx.x;
    const int wave = tid >> 5;
    const int lane = tid & 31;
    const bool hi  = lane >= 16;
    const int  m   = lane & 15;

    const float w0 = w[0], w1 = w[1], w2 = w[2], w3 = w[3];
    const float w4 = w[4], w5 = w[5], w6 = w[6], w7 = w[7];

    float local = 0.f;

#pragma unroll
    for (int it = 0; it < 4; ++it) {
        const int base = (wave * 4 + it) * 16;          // first object of tile
        const float* row = dcp + (size_t)(base + m) * C_DET;

        v2f a0, a1, b0, b1;
        if (!hi) {
            a0.x = row[0]; a0.y = row[1];               // K=0,1
            a1.x = row[4]; a1.y = row[5];               // K=4,5 (second WMMA)
            b0.x = w0;     b0.y = w1;
            b1.x = w4;     b1.y = w5;
        } else {
            a0.x = row[2]; a0.y = row[3];               // K=2,3
            a1.x = row[6]; a1.y = row[7];               // K=6,7
            b0.x = w2;     b0.y = w3;
            b1.x = w6;     b1.y = w7;
        }

        v8f c = {};
        // D = A(K0..3) x B(K0..3) + 0
        c = __builtin_amdgcn_wmma_f32_16x16x4_f32(
                false, a0, false, b0, (short)0, c, false, false);
        // D += A(K4..7) x B(K4..7)
        c = __builtin_amdgcn_wmma_f32_16x16x4_f32(
                false, a1, false, b1, (short)0, c, false, false);

        // Column N=0 of D holds the 16 probs for this tile.
        if (lane == 0 || lane == 16) {
#pragma unroll
            for (int v = 0; v < 8; ++v) {
                float p = c[v];
                local += -fmaxf(logf(p), -100.0f);
            }
        }
    }

    __shared__ float red[512];
    red[tid] = local;
    __syncthreads();
    for (int off = 256; off > 0; off >>= 1) {
        if (tid < off) red[tid] += red[tid + off];
        __syncthreads();
    }
    if (tid == 0) out[0] = red[0] * (1.0f / (float)N_OBJ);
}

// ---------------------------------------------------------------------------
extern "C" void kernel_launch(void* const* d_in, const int* in_sizes, int n_in,
                              void* d_out, int out_size, void* d_ws, size_t ws_size,
                              hipStream_t stream) {
    const float* det_class_probs = (const float*)d_in[0];   // (1024, 8) f32
    const float* det_mask_probs  = (const float*)d_in[1];   // (1024, 8, 28, 28) f32
    const float* seg_mask_probs  = (const float*)d_in[2];   // (1024, 8, 4, 28, 28) f32
    const int*   edge_i          = (const int*)d_in[3];     // (E,) int32
    const int*   edge_j          = (const int*)d_in[4];     // (E,) int32
    const int    n_edges         = in_sizes[3];

    float* partials = (float*)d_ws;                 // 5 * NBLK floats (160 KB)
    float* w_buf    = partials + 5 * NBLK;          // 8 floats
    float* out      = (float*)d_out;                // scalar bce

    cm_stage1_reduce<<<NBLK, 256, 0, stream>>>(det_mask_probs, seg_mask_probs, partials);
    cm_stage2_edges<<<1, 256, 0, stream>>>(partials, edge_i, edge_j, n_edges, w_buf);
    cm_stage3_wmma_bce<<<1, 512, 0, stream>>>(det_class_probs, w_buf, out);
}